// S6_31799937860325
// MI455X (gfx1250) — compile-verified
//
#include <hip/hip_runtime.h>
#include <math.h>
#include <stdint.h>

// ---------------------------------------------------------------------------
// Mamba / S6 forward for MI455X (gfx1250), fp32 end-to-end.
//  - GEMMs on matrix pipes via V_WMMA_F32_16X16X4_F32 (workload is
//    bandwidth-bound at 23.3 TB/s, so full fp32 costs nothing).
//  - A/W slabs staged into LDS with GLOBAL_LOAD_ASYNC_TO_LDS_B128
//    (ASYNCcnt path, no VGPR round-trip), fragments read as single
//    conflict-free ds_load_b64s.
// ---------------------------------------------------------------------------

typedef float v2f __attribute__((ext_vector_type(2)));
typedef float v8f __attribute__((ext_vector_type(8)));

#define DMODEL 1024
#define DINNER 2048
#define DSTATE 16
#define DTRANK 64
#define NPROJ  96        // dt_rank + 2*d_state
#define BATCH  4
#define SEQLEN 2048
#define MROWS  (BATCH * SEQLEN)   // 8192

// Generic pointers to __shared__ live in the LDS aperture: low 32 bits are the
// LDS byte offset, which is exactly what the async-to-LDS VDST operand wants.
__device__ __forceinline__ uint32_t lds_off32(const void* p) {
    return (uint32_t)(uintptr_t)p;
}

// memory -> LDS, 16 bytes per lane, tracked by ASYNCcnt (gfx1250)
__device__ __forceinline__ void async_g2lds_b128(uint32_t lds_byte_off,
                                                 const float* gaddr) {
    asm volatile("global_load_async_to_lds_b128 %0, %1, off"
                 :: "v"(lds_byte_off), "v"(gaddr)
                 : "memory");
}
__device__ __forceinline__ void wait_asynccnt0() {
    asm volatile("s_wait_asynccnt 0x0" ::: "memory");
}

// ---------------------------------------------------------------------------
// Generic fp32 WMMA GEMM:  C[M,N] = A[M,K] * W[N,K]^T   (torch Linear layout)
//   block tile 128x128, 256 threads = 8 wave32, each wave owns 32x64.
//   K staged in 16-wide slabs; LDS layout is [g][row][0..3] with g = k/4 so a
//   b128 async lands one aligned float4 and a WMMA fragment is one b64 read.
//   epilogue: 0 = none, 1 = softplus(v + bias[n])
// M multiple of 128, K multiple of 16; N is bounds-checked.
// ---------------------------------------------------------------------------
__global__ __launch_bounds__(256)
void gemm_wmma_f32(const float* __restrict__ A, int lda,
                   const float* __restrict__ W,
                   float* __restrict__ C, int ldc,
                   int M, int N, int K,
                   const float* __restrict__ bias, int epilogue)
{
    __shared__ float sA[4 * 128 * 4];   // [g][row][4]
    __shared__ float sB[4 * 128 * 4];   // [g][nrow][4]

    const int tid  = threadIdx.x;
    const int wave = tid >> 5;
    const int lane = tid & 31;
    const int half = lane >> 4;      // 0: lanes 0-15, 1: lanes 16-31
    const int l16  = lane & 15;
    const int kb2  = 2 * half;       // ISA 16x4 f32: VGPR0=K{0,2}, VGPR1=K{1,3}

    const int bm = blockIdx.y * 128;
    const int bn = blockIdx.x * 128;
    const int wm = (wave & 3) * 32;  // wave row offset in block
    const int wn = (wave >> 2) * 64; // wave col offset in block

    const uint32_t sA0 = lds_off32(&sA[0]);
    const uint32_t sB0 = lds_off32(&sB[0]);

    // per-thread staging slot: idx = tid + i*256 -> row = idx>>2, g = idx&3
    const int r0 = tid >> 2;         // 0..63
    const int g0 = tid & 3;

    v8f acc[2][4];
    const v8f vzero = {0.f, 0.f, 0.f, 0.f, 0.f, 0.f, 0.f, 0.f};
#pragma unroll
    for (int i = 0; i < 2; ++i)
#pragma unroll
        for (int j = 0; j < 4; ++j) acc[i][j] = vzero;

    for (int k0 = 0; k0 < K; k0 += 16) {
        __syncthreads();   // WAR: previous slab fully consumed
        // stage A block: 128 rows x 16 k  (512 float4 = 2 async b128/thread)
#pragma unroll
        for (int i = 0; i < 2; ++i) {
            const int r = r0 + i * 64;
            async_g2lds_b128(sA0 + (uint32_t)(g0 * 128 + r) * 16u,
                             A + (size_t)(bm + r) * lda + k0 + g0 * 4);
        }
        // stage W block: 128 n-rows x 16 k, un-transposed
#pragma unroll
        for (int i = 0; i < 2; ++i) {
            const int n = r0 + i * 64;
            if (bn + n < N)
                async_g2lds_b128(sB0 + (uint32_t)(g0 * 128 + n) * 16u,
                                 W + (size_t)(bn + n) * K + k0 + g0 * 4);
        }
        wait_asynccnt0();
        __syncthreads();   // all waves' slabs visible

#pragma unroll
        for (int ks = 0; ks < 4; ++ks) {
            v2f a0 = *(const v2f*)&sA[(ks * 128 + wm + l16) * 4 + kb2];
            v2f a1 = *(const v2f*)&sA[(ks * 128 + wm + 16 + l16) * 4 + kb2];
#pragma unroll
            for (int nt = 0; nt < 4; ++nt) {
                v2f bf = *(const v2f*)&sB[(ks * 128 + wn + nt * 16 + l16) * 4 + kb2];
                acc[0][nt] = __builtin_amdgcn_wmma_f32_16x16x4_f32(
                    false, a0, false, bf, (short)0, acc[0][nt], false, false);
                acc[1][nt] = __builtin_amdgcn_wmma_f32_16x16x4_f32(
                    false, a1, false, bf, (short)0, acc[1][nt], false, false);
            }
        }
    }

    // C/D layout: lane gives N, VGPR r gives M = r + 8*half
#pragma unroll
    for (int mt = 0; mt < 2; ++mt) {
#pragma unroll
        for (int nt = 0; nt < 4; ++nt) {
            const int n = bn + wn + nt * 16 + l16;
            if (n >= N) continue;
#pragma unroll
            for (int r = 0; r < 8; ++r) {
                const int m = bm + wm + mt * 16 + half * 8 + r;
                float v = acc[mt][nt][r];
                if (epilogue == 1) {            // softplus(v + bias)
                    v += bias[n];
                    v = (v > 20.f) ? v : log1pf(expf(v));
                }
                C[(size_t)m * ldc + n] = v;
            }
        }
    }
}

// ---------------------------------------------------------------------------
// Causal depthwise conv1d (k=4) + SiLU.  xz holds [x_branch | z] with ld 4096.
// ---------------------------------------------------------------------------
__global__ __launch_bounds__(256)
void conv_silu(const float* __restrict__ xz, const float* __restrict__ cw,
               const float* __restrict__ cb, float* __restrict__ xcv)
{
    const size_t i   = (size_t)blockIdx.x * 256 + threadIdx.x;  // b*L*Di total
    const int    c   = (int)(i & (DINNER - 1));
    const size_t row = i >> 11;                 // b*L + l
    const int    l   = (int)(row & (SEQLEN - 1));
    float acc = cb[c];
#pragma unroll
    for (int j = 0; j < 4; ++j) {
        const int ll = l - 3 + j;
        if (ll >= 0)
            acc += cw[c * 4 + j] * xz[(row - 3 + j) * (size_t)(2 * DINNER) + c];
    }
    xcv[i] = acc / (1.f + expf(-acc));          // SiLU
}

// ---------------------------------------------------------------------------
// Selective scan, fused with +u*D and the SiLU(z) gate.
// One wave32 handles 2 (b,d) channels; lane = state index n (16 per channel).
// Loads for step t+1 are issued while step t's exp/FMA chain executes.
// y is written in place over x_conv (u[t] is read before y[t] is stored).
// ---------------------------------------------------------------------------
__global__ __launch_bounds__(256)
void selective_scan(const float* __restrict__ xz,    // z at col DINNER+d, ld 4096
                    const float* __restrict__ xcv,   // u  [M, DINNER]
                    const float* __restrict__ xdbl,  // B at 64+n, C at 80+n, ld 96
                    const float* __restrict__ dt,    // [M, DINNER]
                    const float* __restrict__ A_log, // [DINNER, 16]
                    const float* __restrict__ Dp,    // [DINNER]
                    float* __restrict__ y)           // == xcv (in place)
{
    const int gwave = (int)((blockIdx.x * 256 + threadIdx.x) >> 5); // 0..4095
    const int lane  = threadIdx.x & 31;
    const int half  = lane >> 4;
    const int n     = lane & 15;
    const int ch    = gwave * 2 + half;          // b*DINNER + d, 0..8191
    const int b     = ch >> 11;
    const int d     = ch & (DINNER - 1);

    const float Ad = -expf(A_log[d * DSTATE + n]);
    const float Dd = Dp[d];
    float h = 0.f;
    const size_t base = (size_t)b * SEQLEN;

    // prologue: loads for t = 0
    float dtv = dt  [base * DINNER + d];
    float u   = xcv [base * DINNER + d];
    float Bn  = xdbl[base * NPROJ + DTRANK + n];
    float Cn  = xdbl[base * NPROJ + DTRANK + DSTATE + n];
    float z   = xz  [base * (size_t)(2 * DINNER) + DINNER + d];

    for (int t = 0; t < SEQLEN; ++t) {
        // issue next step's loads early (independent of the recurrence)
        float dtv_n = 0.f, u_n = 0.f, Bn_n = 0.f, Cn_n = 0.f, z_n = 0.f;
        if (t + 1 < SEQLEN) {
            const size_t rn = base + t + 1;
            dtv_n = dt  [rn * DINNER + d];
            u_n   = xcv [rn * DINNER + d];
            Bn_n  = xdbl[rn * NPROJ + DTRANK + n];
            Cn_n  = xdbl[rn * NPROJ + DTRANK + DSTATE + n];
            z_n   = xz  [rn * (size_t)(2 * DINNER) + DINNER + d];
        }

        h = expf(dtv * Ad) * h + (dtv * u) * Bn;
        float yv = h * Cn;
        // reduce over the 16 states (stays inside the 16-lane half)
        yv += __shfl_xor(yv, 1);
        yv += __shfl_xor(yv, 2);
        yv += __shfl_xor(yv, 4);
        yv += __shfl_xor(yv, 8);

        if (n == 0) {
            const float g = z / (1.f + expf(-z));       // SiLU gate
            y[(base + t) * DINNER + d] = (yv + u * Dd) * g;
        }

        dtv = dtv_n; u = u_n; Bn = Bn_n; Cn = Cn_n; z = z_n;
    }
}

// ---------------------------------------------------------------------------
extern "C" void kernel_launch(void* const* d_in, const int* in_sizes, int n_in,
                              void* d_out, int out_size, void* d_ws, size_t ws_size,
                              hipStream_t stream)
{
    (void)in_sizes; (void)n_in; (void)out_size; (void)ws_size;
    const float* x         = (const float*)d_in[0];
    const float* in_proj_w = (const float*)d_in[1];   // [2*Di, Dm]
    const float* conv_w    = (const float*)d_in[2];   // [Di, 1, 4]
    const float* conv_b    = (const float*)d_in[3];   // [Di]
    const float* x_proj_w  = (const float*)d_in[4];   // [96, Di]
    const float* dt_proj_w = (const float*)d_in[5];   // [Di, 64]
    const float* dt_proj_b = (const float*)d_in[6];   // [Di]
    const float* A_log     = (const float*)d_in[7];   // [Di, 16]
    const float* D_param   = (const float*)d_in[8];   // [Di]
    const float* out_w     = (const float*)d_in[9];   // [Dm, Di]
    float* out = (float*)d_out;

    // workspace layout (fp32):
    float* xz   = (float*)d_ws;                         // [M, 2*Di]  128 MB
    float* xcv  = xz   + (size_t)MROWS * (2 * DINNER);  // [M, Di]     64 MB (u, then y in place)
    float* xdbl = xcv  + (size_t)MROWS * DINNER;        // [M, 96]      3 MB
    float* dtb  = xdbl + (size_t)MROWS * NPROJ;         // [M, Di]     64 MB

    const dim3 blk(256);

    // 1) in_proj: xz = x @ W_in^T           (M=8192, N=4096, K=1024)
    gemm_wmma_f32<<<dim3(4096 / 128, MROWS / 128), blk, 0, stream>>>(
        x, DMODEL, in_proj_w, xz, 2 * DINNER, MROWS, 2 * DINNER, DMODEL,
        nullptr, 0);

    // 2) causal depthwise conv + SiLU -> xcv
    conv_silu<<<(MROWS * DINNER) / 256, blk, 0, stream>>>(xz, conv_w, conv_b, xcv);

    // 3) x_proj: xdbl = xcv @ W_xp^T        (N=96, K=2048)
    gemm_wmma_f32<<<dim3(1, MROWS / 128), blk, 0, stream>>>(
        xcv, DINNER, x_proj_w, xdbl, NPROJ, MROWS, NPROJ, DINNER,
        nullptr, 0);

    // 4) dt = softplus(dt_lr @ W_dt^T + b)  (A = xdbl cols 0..63, N=2048, K=64)
    gemm_wmma_f32<<<dim3(DINNER / 128, MROWS / 128), blk, 0, stream>>>(
        xdbl, NPROJ, dt_proj_w, dtb, DINNER, MROWS, DINNER, DTRANK,
        dt_proj_b, 1);

    // 5) selective scan + D-skip + SiLU(z) gate, y overwrites xcv
    selective_scan<<<(BATCH * DINNER / 2) / 8, blk, 0, stream>>>(
        xz, xcv, xdbl, dtb, A_log, D_param, xcv);

    // 6) out_proj: out = y @ W_out^T        (N=1024, K=2048)
    gemm_wmma_f32<<<dim3(DMODEL / 128, MROWS / 128), blk, 0, stream>>>(
        xcv, DINNER, out_w, out, DMODEL, MROWS, DMODEL, DINNER,
        nullptr, 0);
}